// ConfusionDropout_52407190946399
// MI455X (gfx1250) — compile-verified
//
#include <hip/hip_runtime.h>
#include <stdint.h>

#define DIM   4096      // channels per row
#define NCLS  1000      // classes
#define KDROP 1024u     // DIM * 0.25
#define NT    256       // threads per block (8 wave32 waves)

typedef __attribute__((ext_vector_type(4))) float fvec4;

struct Top2 { float v1, v2; int i1, i2; };

__device__ __forceinline__ void top2_insert(Top2& a, float v, int i) {
    if (v > a.v1 || (v == a.v1 && i < a.i1)) {
        a.v2 = a.v1; a.i2 = a.i1; a.v1 = v; a.i1 = i;
    } else if (v > a.v2 || (v == a.v2 && i < a.i2)) {
        a.v2 = v; a.i2 = i;
    }
}

__global__ __launch_bounds__(NT) void confusion_dropout_kernel(
    const float* __restrict__ x,
    const float* __restrict__ po,
    const float* __restrict__ W,
    float* __restrict__ out)
{
    const int t    = threadIdx.x;
    const int b    = blockIdx.x;
    const int lane = t & 31;
    const int wv   = t >> 5;

    __shared__ fvec4    s_x4[DIM / 4];   // 16 KB: x row
    __shared__ fvec4    s_sc4[DIM / 4];  // 16 KB: scores
    __shared__ unsigned s_hist[NT];      // 256 radix bins (bin t owned by thread t)
    __shared__ unsigned s_wred[8];       // per-wave partials
    __shared__ Top2     s_t2[8];         // per-wave top-2 winners
    __shared__ int      s_i0, s_i1;      // winning class indices
    __shared__ unsigned s_selbin, s_selk;

    // ---- 1) async-copy the x row (16 KB) global -> LDS, 4 x B128 per lane ----
    {
        const char* grow = (const char*)(x + (size_t)b * DIM);
        #pragma unroll
        for (int k = 0; k < 4; ++k) {
            int f = t + k * NT;
            unsigned lds = (unsigned)(uintptr_t)(&s_x4[f]);
            unsigned long long ga = (unsigned long long)(uintptr_t)(grow + (size_t)f * 16);
            asm volatile("global_load_async_to_lds_b128 %0, %1, off"
                         :: "v"(lds), "v"(ga) : "memory");
        }
    }

    // ---- 2) top-2 of prev_output (overlaps async copy); wave32 butterfly ----
    Top2 tp; tp.v1 = tp.v2 = -__builtin_inff(); tp.i1 = tp.i2 = 0x7fffffff;
    {
        const float* prow = po + (size_t)b * NCLS;
        for (int j = t; j < NCLS; j += NT)
            top2_insert(tp, __builtin_nontemporal_load(prow + j), j);
    }
    #pragma unroll
    for (int off = 1; off < 32; off <<= 1) {
        float ov1 = __shfl_xor(tp.v1, off, 32);
        int   oi1 = __shfl_xor(tp.i1, off, 32);
        float ov2 = __shfl_xor(tp.v2, off, 32);
        int   oi2 = __shfl_xor(tp.i2, off, 32);
        top2_insert(tp, ov1, oi1);
        top2_insert(tp, ov2, oi2);
    }
    if (lane == 0) s_t2[wv] = tp;
    s_hist[t] = 0;                       // clear bins for the fused byte-3 pass

    asm volatile("s_wait_asynccnt 0" ::: "memory");
    __syncthreads();

    // ---- 3) merge the 8 wave winners on wave 0 (3-step butterfly) ----
    if (wv == 0) {
        Top2 m;
        if (lane < 8) m = s_t2[lane];
        else { m.v1 = m.v2 = -__builtin_inff(); m.i1 = m.i2 = 0x7fffffff; }
        #pragma unroll
        for (int off = 1; off < 8; off <<= 1) {
            float ov1 = __shfl_xor(m.v1, off, 32);
            int   oi1 = __shfl_xor(m.i1, off, 32);
            float ov2 = __shfl_xor(m.v2, off, 32);
            int   oi2 = __shfl_xor(m.i2, off, 32);
            top2_insert(m, ov1, oi1);
            top2_insert(m, ov2, oi2);
        }
        if (lane == 0) { s_i0 = m.i1; s_i1 = m.i2; }
    }
    __syncthreads();

    // ---- 4) scores = |x * (W[i0]-W[i1])| (coalesced) + fused byte-3 histogram -
    {
        const fvec4* W0 = (const fvec4*)(W + (size_t)s_i0 * DIM);
        const fvec4* W1 = (const fvec4*)(W + (size_t)s_i1 * DIM);
        #pragma unroll
        for (int k = 0; k < 4; ++k) {
            int f = t + k * NT;
            fvec4 xv = s_x4[f];
            fvec4 a = W0[f], c = W1[f];
            fvec4 sc;
            sc.x = __builtin_fabsf(xv.x * (a.x - c.x));
            sc.y = __builtin_fabsf(xv.y * (a.y - c.y));
            sc.z = __builtin_fabsf(xv.z * (a.z - c.z));
            sc.w = __builtin_fabsf(xv.w * (a.w - c.w));
            s_sc4[f] = sc;
            atomicAdd(&s_hist[__float_as_uint(sc.x) >> 24], 1u);
            atomicAdd(&s_hist[__float_as_uint(sc.y) >> 24], 1u);
            atomicAdd(&s_hist[__float_as_uint(sc.z) >> 24], 1u);
            atomicAdd(&s_hist[__float_as_uint(sc.w) >> 24], 1u);
        }
    }
    __syncthreads();

    // ---- 5) this thread's contiguous 16 scores as monotone uint bits ----
    unsigned u[16];
    #pragma unroll
    for (int i = 0; i < 4; ++i) {
        fvec4 v = s_sc4[4 * t + i];
        u[4 * i + 0] = __float_as_uint(v.x);
        u[4 * i + 1] = __float_as_uint(v.y);
        u[4 * i + 2] = __float_as_uint(v.z);
        u[4 * i + 3] = __float_as_uint(v.w);
    }

    // ---- 6) byte radix-select of the KDROP-th largest (4 passes, pass 3 fused)
    unsigned kk = KDROP;     // residual rank within the current prefix group
    unsigned prefix = 0;

    auto select_pass = [&](int p) {
        unsigned own = s_hist[t];
        unsigned inc = own;                       // suffix-inclusive within wave
        #pragma unroll
        for (int off = 1; off < 32; off <<= 1) {
            unsigned o = __shfl_down(inc, off, 32);
            if (lane + off < 32) inc += o;
        }
        if (lane == 0) s_wred[wv] = inc;          // wave (bin-group) total
        __syncthreads();
        unsigned H = 0;
        #pragma unroll
        for (int w = 0; w < 8; ++w) if (w > wv) H += s_wred[w];
        unsigned g_excl = H + inc - own;          // count in strictly higher bins
        unsigned g_incl = H + inc;
        if (g_excl < kk && kk <= g_incl) {        // exactly one thread fires
            s_selbin = (unsigned)t;
            s_selk   = kk - g_excl;
        }
        __syncthreads();
        prefix |= s_selbin << (8 * p);
        kk = s_selk;
    };

    select_pass(3);                               // histogram came fused above

    #pragma unroll
    for (int p = 2; p >= 0; --p) {
        s_hist[t] = 0;
        __syncthreads();
        unsigned mask = 0xFFFFFFFFu << (8 * (p + 1));
        #pragma unroll
        for (int i = 0; i < 16; ++i) {
            if ((u[i] & mask) == prefix)
                atomicAdd(&s_hist[(u[i] >> (8 * p)) & 0xFFu], 1u);
        }
        __syncthreads();
        select_pass(p);
    }

    const unsigned T    = prefix;  // bit pattern of the KDROP-th largest score
    const unsigned need = kk;      // # of threshold-equal elements to drop

    // ---- 7) index-ordered rank among equals (wave scan + wave combine) ----
    unsigned ceq = 0;
    #pragma unroll
    for (int i = 0; i < 16; ++i) ceq += (u[i] == T) ? 1u : 0u;
    unsigned incp = ceq;
    #pragma unroll
    for (int off = 1; off < 32; off <<= 1) {
        unsigned o = __shfl_up(incp, off, 32);
        if (lane >= off) incp += o;
    }
    if (lane == 31) s_wred[wv] = incp;
    __syncthreads();
    unsigned base = incp - ceq;                   // exclusive within wave
    #pragma unroll
    for (int w = 0; w < 8; ++w) if (w < wv) base += s_wred[w];

    // ---- 8) drop: all > T, plus first `need` equal-to-T in channel order ----
    {
        float* sx = (float*)s_x4;
        int eqrank = (int)base;
        #pragma unroll
        for (int i = 0; i < 16; ++i) {
            int j = 16 * t + i;
            bool drop;
            if (u[i] > T)       drop = true;
            else if (u[i] == T) { drop = (eqrank < (int)need); ++eqrank; }
            else                drop = false;
            if (drop) sx[j] = sx[j] * 0.0f;       // x*0 keeps JAX's +/-0 bits
        }
    }
    __syncthreads();

    // ---- 9) coalesced non-temporal store of the masked row ----
    {
        fvec4* out4 = (fvec4*)(out + (size_t)b * DIM);
        #pragma unroll
        for (int k = 0; k < 4; ++k) {
            int f = t + k * NT;
            __builtin_nontemporal_store(s_x4[f], &out4[f]);
        }
    }
}

extern "C" void kernel_launch(void* const* d_in, const int* in_sizes, int n_in,
                              void* d_out, int out_size, void* d_ws, size_t ws_size,
                              hipStream_t stream) {
    (void)n_in; (void)out_size; (void)d_ws; (void)ws_size;
    const float* x  = (const float*)d_in[0];
    const float* po = (const float*)d_in[1];
    const float* W  = (const float*)d_in[2];
    float* out = (float*)d_out;
    const int B = in_sizes[0] / DIM;   // 16384
    confusion_dropout_kernel<<<dim3(B), dim3(NT), 0, stream>>>(x, po, W, out);
}